// RECT_L_14310831030635
// MI455X (gfx1250) — compile-verified
//
#include <hip/hip_runtime.h>

typedef __attribute__((ext_vector_type(2))) float v2f;
typedef __attribute__((ext_vector_type(4))) float v4f;
typedef __attribute__((ext_vector_type(8))) float v8f;
typedef unsigned int u32x4 __attribute__((ext_vector_type(4)));
typedef int          i32x4 __attribute__((ext_vector_type(4)));
typedef int          i32x8 __attribute__((ext_vector_type(8)));

#define N_NODES   100000
#define E_EDGES   800000
#define IN_FEATS  512
#define N_HIDDEN  128

// LDS row stride for the A tile: 512 floats + TDM pad (2 DWORDs per 256 DWORDs)
#define LDS_ROW   516

// Detect toolchain flavor for the TDM builtin arity (therock headers => 6-arg).
#if defined(__has_include)
#  if __has_include(<hip/amd_detail/amd_gfx1250_TDM.h>)
#    define GCN_TDM_6ARG 1
#  endif
#endif

// ---------------------------------------------------------------------------
// Repack W_gcn [512,128] into K-pair-interleaved form so the GEMM1 B operand
// (K=k0+koff, K+1 at fixed n) is one contiguous b64 load:
//   Wp[((k>>1)*128 + n)*2 + (k&1)] = W[k*128 + n]
// ---------------------------------------------------------------------------
__global__ __launch_bounds__(256)
void gcn_repack_w(const float* __restrict__ W, float* __restrict__ Wp)
{
    const int i = blockIdx.x * 256 + threadIdx.x;     // 256 blocks * 256 = 512*128
    const int k = i >> 7;
    const int n = i & 127;
    Wp[(size_t)((k >> 1) * N_HIDDEN + n) * 2 + (k & 1)] = W[i];
}

// ---------------------------------------------------------------------------
// GEMM1: X[N,128] = A[N,512] @ W[512,128]   (fp32, V_WMMA_F32_16X16X4_F32)
// Block = 256 threads = 8 waves = one 16-row m-strip; wave w owns n-tile w.
// A tile (16x512, 32KB) staged in LDS by the Tensor Data Mover; operands read
// with conflict-free ds_load_b64 (row stride 516 floats -> banks 4r..4r+3).
// ---------------------------------------------------------------------------
__global__ __launch_bounds__(256)
void gcn_gemm1_f32wmma(const float* __restrict__ A,
                       const float* __restrict__ Wp,
                       float* __restrict__ X)
{
    __shared__ float ldsA[16 * LDS_ROW];              // 33 KB

    const int lane = threadIdx.x & 31;
    const int wave = threadIdx.x >> 5;
    const int m0   = blockIdx.x * 16;                 // 6250 blocks
    const int half = lane >> 4;
    const int lid  = lane & 15;
    const int koff = half * 2;

#if __has_builtin(__builtin_amdgcn_tensor_load_to_lds)
    if (wave == 0) {
        const unsigned long long ga =
            (unsigned long long)(const void*)(A + (size_t)m0 * IN_FEATS);
        const unsigned ldsoff =
            (unsigned)(unsigned long long)(const void*)&ldsA[0];

        // D# group 0: count=1 | lds_addr | global_addr(57b) | type=2
        u32x4 g0;
        g0[0] = 1u;                                   // count = 1 valid descriptor
        g0[1] = ldsoff;                               // LDS byte address
        g0[2] = (unsigned)ga;                         // global addr [31:0]
        g0[3] = ((unsigned)(ga >> 32) & 0x01FFFFFFu)  // global addr [56:32]
                | (2u << 30);                         // type = 2 ("image")
        // D# group 1: data_size=4B, pad 2 DWORDs every 256 DWORDs,
        // tensor [512 x N], tile [512 x 16], dim0 stride 512.
        i32x8 g1;
        g1[0] = (int)((2u << 16)                      // data_size = 4 bytes
                    | (1u << 20)                      // pad_enable
                    | (7u << 22)                      // pad_interval: 256 DWORDs
                    | (1u << 25));                    // pad_amount: 2 DWORDs
        g1[1] = (int)((unsigned)IN_FEATS << 16);      // tensor_dim0[15:0] = 512
        g1[2] = (int)(((unsigned)N_NODES & 0xFFFFu) << 16);  // dim0 hi=0 | dim1 lo
        g1[3] = (int)(((unsigned)N_NODES >> 16)       // tensor_dim1[31:16]
                    | ((unsigned)IN_FEATS << 16));    // tile_dim0 = 512
        g1[4] = 16;                                   // tile_dim1 = 16 rows
        g1[5] = IN_FEATS;                             // tensor_dim0_stride = 512
        g1[6] = 0;
        g1[7] = 0;
        const i32x4 gz = {};
#if defined(GCN_TDM_6ARG)
        const i32x8 gz8 = {};
        __builtin_amdgcn_tensor_load_to_lds(g0, g1, gz, gz, gz8, 0);
#else
        __builtin_amdgcn_tensor_load_to_lds(g0, g1, gz, gz, 0);
#endif
        __builtin_amdgcn_s_wait_tensorcnt(0);
    }
#else
    // Fallback: cooperative copy into the same padded LDS layout (8B chunks).
    for (int it = 0; it < 16; ++it) {
        const int x2  = (it * 256 + threadIdx.x) * 2;
        const int row = x2 >> 9;
        const int k   = x2 & 511;
        *(v2f*)(ldsA + LDS_ROW * row + k + (k >= 256 ? 2 : 0)) =
            *(const v2f*)(A + (size_t)(m0 + row) * IN_FEATS + k);
    }
#endif
    __syncthreads();

    const int n0 = wave * 16;                         // 8 n-tiles of 16
    v8f acc = {};
    const float* arow = ldsA + LDS_ROW * lid + koff;  // A[m][k] from LDS
    const float* bp   = Wp + (size_t)half * 256 + (size_t)(n0 + lid) * 2;

    #pragma unroll 8
    for (int k0 = 0; k0 < 256; k0 += 4) {             // K < 256 (no pad skew)
        v2f a = *(const v2f*)(arow + k0);
        v2f b = *(const v2f*)(bp + (size_t)k0 * N_HIDDEN);
        acc = __builtin_amdgcn_wmma_f32_16x16x4_f32(
                  false, a, false, b, (short)0, acc, false, false);
    }
    #pragma unroll 8
    for (int k0 = 256; k0 < IN_FEATS; k0 += 4) {      // K >= 256 (+2 DWORD pad)
        v2f a = *(const v2f*)(arow + k0 + 2);
        v2f b = *(const v2f*)(bp + (size_t)k0 * N_HIDDEN);
        acc = __builtin_amdgcn_wmma_f32_16x16x4_f32(
                  false, a, false, b, (short)0, acc, false, false);
    }

    float* xp = X + (size_t)(m0 + 8 * half) * N_HIDDEN + n0 + lid;
    #pragma unroll
    for (int v = 0; v < 8; ++v)
        xp[(size_t)v * N_HIDDEN] = acc[v];
}

// ---------------------------------------------------------------------------
// Init: H[n][j] = b_gcn[j]
// ---------------------------------------------------------------------------
__global__ __launch_bounds__(256)
void gcn_init_h(float* __restrict__ H, const float* __restrict__ b)
{
    const int i = blockIdx.x * 256 + threadIdx.x;     // 50000 blocks
    H[i] = b[i & (N_HIDDEN - 1)];
}

// ---------------------------------------------------------------------------
// Edge scatter: H[dst] += edge_attr[e] * X[src]; one wave per edge,
// float4 per lane, f32 near-atomics in L2 (X and H are both L2-resident).
// ---------------------------------------------------------------------------
__global__ __launch_bounds__(256)
void gcn_scatter_edges(const float* __restrict__ X,
                       const long long* __restrict__ edge_index,
                       const float* __restrict__ edge_attr,
                       float* __restrict__ H)
{
    const int lane = threadIdx.x & 31;
    const int wave = threadIdx.x >> 5;
    const int e    = blockIdx.x * 8 + wave;           // 100000 blocks * 8 = E
    const long long src = edge_index[e];
    const long long dst = edge_index[E_EDGES + e];
    const float w = edge_attr[e];

    v4f m = *(const v4f*)(X + (size_t)src * N_HIDDEN + lane * 4);
    m *= w;
    float* hp = H + (size_t)dst * N_HIDDEN + lane * 4;
    unsafeAtomicAdd(hp + 0, m.x);
    unsafeAtomicAdd(hp + 1, m.y);
    unsafeAtomicAdd(hp + 2, m.z);
    unsafeAtomicAdd(hp + 3, m.w);
}

// ---------------------------------------------------------------------------
// GEMM2: OUT[N,512] = H[N,128] @ W_fc^T + b_fc
// Two n-tiles per wave: one A b64 load feeds two WMMAs (16 acc VGPRs).
// ---------------------------------------------------------------------------
__global__ __launch_bounds__(256)
void gcn_gemm2_f32wmma(const float* __restrict__ H,
                       const float* __restrict__ Wfc,
                       const float* __restrict__ bfc,
                       float* __restrict__ OUT)
{
    const int lane = threadIdx.x & 31;
    const int wave = threadIdx.x >> 5;
    const int pair = blockIdx.x * 8 + wave;           // 12500 blocks * 8 = 100000
    const int tn   = pair & 15;                       // 16 tile-pairs across 512
    const int tm   = pair >> 4;                       // 6250 m-tiles
    const int m0   = tm * 16;
    const int n0   = tn * 32;
    const int half = lane >> 4;
    const int lid  = lane & 15;
    const int koff = half * 2;

    v8f acc0 = {}, acc1 = {};
    const float* ap  = H   + (size_t)(m0 + lid) * N_HIDDEN + koff;
    const float* bp0 = Wfc + (size_t)(n0 + lid) * N_HIDDEN + koff;      // B[k][n]
    const float* bp1 = bp0 + (size_t)16 * N_HIDDEN;

    #pragma unroll 8
    for (int k0 = 0; k0 < N_HIDDEN; k0 += 4) {
        v2f a  = *(const v2f*)(ap  + k0);
        v2f b0 = *(const v2f*)(bp0 + k0);
        v2f b1 = *(const v2f*)(bp1 + k0);
        acc0 = __builtin_amdgcn_wmma_f32_16x16x4_f32(
                   false, a, false, b0, (short)0, acc0, false, false);
        acc1 = __builtin_amdgcn_wmma_f32_16x16x4_f32(
                   false, a, false, b1, (short)0, acc1, false, false);
    }

    const float bias0 = bfc[n0 + lid];
    const float bias1 = bfc[n0 + 16 + lid];
    float* op = OUT + (size_t)(m0 + 8 * half) * IN_FEATS + n0 + lid;
    #pragma unroll
    for (int v = 0; v < 8; ++v) {
        op[(size_t)v * IN_FEATS]      = acc0[v] + bias0;
        op[(size_t)v * IN_FEATS + 16] = acc1[v] + bias1;
    }
}

// ---------------------------------------------------------------------------
extern "C" void kernel_launch(void* const* d_in, const int* in_sizes, int n_in,
                              void* d_out, int out_size, void* d_ws, size_t ws_size,
                              hipStream_t stream)
{
    const float*     inputs     = (const float*)d_in[0];
    const long long* edge_index = (const long long*)d_in[1];   // int64 per reference
    const float*     edge_attr  = (const float*)d_in[2];
    const float*     W_gcn      = (const float*)d_in[3];
    const float*     b_gcn      = (const float*)d_in[4];
    const float*     W_fc       = (const float*)d_in[5];
    const float*     b_fc       = (const float*)d_in[6];
    float*           out        = (float*)d_out;

    // Workspace: X [N,128] | H [N,128] | Wp [512*128] (fp32)
    float* X  = (float*)d_ws;
    float* H  = X + (size_t)N_NODES * N_HIDDEN;
    float* Wp = H + (size_t)N_NODES * N_HIDDEN;

    gcn_repack_w<<<(IN_FEATS * N_HIDDEN) / 256, 256, 0, stream>>>(W_gcn, Wp);
    gcn_gemm1_f32wmma<<<N_NODES / 16, 256, 0, stream>>>(inputs, Wp, X);
    gcn_init_h<<<(N_NODES * N_HIDDEN) / 256, 256, 0, stream>>>(H, b_gcn);
    gcn_scatter_edges<<<E_EDGES / 8, 256, 0, stream>>>(X, edge_index, edge_attr, H);
    gcn_gemm2_f32wmma<<<(N_NODES / 16) * (IN_FEATS / 16) / 16, 256, 0, stream>>>(
        H, W_fc, b_fc, out);
}